// TripletLossBrdfSim_48223892799927
// MI455X (gfx1250) — compile-verified
//
#include <hip/hip_runtime.h>
#include <hip/hip_bf16.h>
#include <math.h>

typedef __attribute__((ext_vector_type(2))) float v2f;
typedef __attribute__((ext_vector_type(8))) float v8f;

#define B_ROWS 8192
#define DIMS   128
#define NCLS   100
#define MARGIN 0.3f

// ---------------------------------------------------------------------------
// Kernel 1: per-row squared norms sq[b] = ||x_b||^2  (one 4MB sweep),
// and zero the scalar loss accumulator (thread 0).
// ---------------------------------------------------------------------------
__global__ void row_norms_kernel(const float* __restrict__ x,
                                 float* __restrict__ sq,
                                 float* __restrict__ out) {
    int r = blockIdx.x * blockDim.x + threadIdx.x;
    if (r == 0) out[0] = 0.0f;
    if (r >= B_ROWS) return;
    const float4* p = (const float4*)(x + (size_t)r * DIMS);
    float s = 0.0f;
#pragma unroll
    for (int i = 0; i < DIMS / 4; ++i) {
        float4 v = p[i];
        s += v.x * v.x + v.y * v.y + v.z * v.z + v.w * v.w;
    }
    sq[r] = s;
}

// ---------------------------------------------------------------------------
// Kernel 2: one wave (32 threads) per 16-row tile.
//   A  = x[t*16 .. t*16+15]            (16 x 128, f32)
//   B0 = gathered rows x[i0[...]]      (128 x 16, column per output row)
//   B1 = gathered rows x[i1[...]]
// 32 chained V_WMMA_F32_16X16X4_F32 per accumulator; diag(C) = gathered dots.
//
// f32 16x16x4 fragment layout (ISA 7.12.2):
//   A: lane L -> M = L&15;  VGPR0 = K = 2*(L>>4), VGPR1 = K+1
//   B: lane L -> N = L&15;  VGPR0 = K = 2*(L>>4), VGPR1 = K+1
//   C: lane L, VGPR v -> M = v + 8*(L>>4), N = L&15
// ---------------------------------------------------------------------------
__global__ void __launch_bounds__(32)
triplet_wmma_kernel(const float* __restrict__ x,
                    const int*   __restrict__ targets,
                    const int*   __restrict__ ixs,      // (B,2) row-major
                    const float* __restrict__ brdf,     // (C,C)
                    const float* __restrict__ sq,
                    float* __restrict__ out) {
    const int t    = blockIdx.x;     // tile index (16 rows)
    const int L    = threadIdx.x;    // 0..31
    const int col  = L & 15;
    const int half = L >> 4;
    const int kb   = half * 2;       // this lane's K sub-offset {0,2}

    const int rowA = t * 16 + col;           // A row / output row for this lane
    const int g0   = ixs[rowA * 2 + 0];      // gathered row indices
    const int g1   = ixs[rowA * 2 + 1];

    const float* pa  = x + (size_t)rowA * DIMS + kb;
    const float* pb0 = x + (size_t)g0   * DIMS + kb;
    const float* pb1 = x + (size_t)g1   * DIMS + kb;

    v8f c0 = {};
    v8f c1 = {};
#pragma unroll
    for (int k = 0; k < DIMS; k += 4) {
        v2f a  = *(const v2f*)(pa  + k);   // A[M=col][k+kb : k+kb+1]
        v2f b0 = *(const v2f*)(pb0 + k);   // B0[k+kb : k+kb+1][N=col]
        v2f b1 = *(const v2f*)(pb1 + k);
        // 8 args: (neg_a, A, neg_b, B, c_mod, C, reuse_a, reuse_b)
        c0 = __builtin_amdgcn_wmma_f32_16x16x4_f32(false, a, false, b0,
                                                   (short)0, c0, false, false);
        c1 = __builtin_amdgcn_wmma_f32_16x16x4_f32(false, a, false, b1,
                                                   (short)0, c1, false, false);
    }

    // Extract diag(C): element i lives in lane (i<8 ? i : 16+i) at VGPR (i&7).
    __shared__ float diag0[16];
    __shared__ float diag1[16];
#pragma unroll
    for (int v = 0; v < 8; ++v) {
        int M = v + 8 * half;
        if (M == col) {                  // each diag slot written by exactly 1 lane
            diag0[col] = c0[v];
            diag1[col] = c1[v];
        }
    }
    __syncthreads();

    __shared__ float red[16];
    if (L < 16) {
        float sqa  = sq[rowA];
        float d2_0 = sqa + sq[g0] - 2.0f * diag0[L];
        float d2_1 = sqa + sq[g1] - 2.0f * diag1[L];
        float d0 = sqrtf(fmaxf(d2_0, 1e-12f));   // matches clip(d2, 1e-12) + sqrt
        float d1 = sqrtf(fmaxf(d2_1, 1e-12f));

        int   ta  = targets[rowA];
        float md1 = brdf[ta * NCLS + targets[g0]];
        float md2 = brdf[ta * NCLS + targets[g1]];

        bool  cond = md1 < md2;
        float ap = cond ? d0 : d1;
        float an = cond ? d1 : d0;
        red[L] = fmaxf(ap - an + MARGIN, 0.0f) * (1.0f / (float)B_ROWS);
    }
    __syncthreads();

    if (L == 0) {
        float s = 0.0f;
#pragma unroll
        for (int i = 0; i < 16; ++i) s += red[i];
        atomicAdd(out, s);
    }
}

// ---------------------------------------------------------------------------
// Inputs (setup_inputs order):
//   d_in[0] inputs            f32  [8192*128]
//   d_in[1] targets           int  [8192]
//   d_in[2] ixs               int  [8192*2]
//   d_in[3] brdf_metric_dist  f32  [100*100]
// d_out: 1 float (scalar loss).  d_ws: >= 8192*4 bytes for row norms.
// ---------------------------------------------------------------------------
extern "C" void kernel_launch(void* const* d_in, const int* in_sizes, int n_in,
                              void* d_out, int out_size, void* d_ws, size_t ws_size,
                              hipStream_t stream) {
    const float* x       = (const float*)d_in[0];
    const int*   targets = (const int*)d_in[1];
    const int*   ixs     = (const int*)d_in[2];
    const float* brdf    = (const float*)d_in[3];
    float*       out     = (float*)d_out;
    float*       sq      = (float*)d_ws;

    row_norms_kernel<<<B_ROWS / 256, 256, 0, stream>>>(x, sq, out);
    triplet_wmma_kernel<<<B_ROWS / 16, 32, 0, stream>>>(x, targets, ixs, brdf, sq, out);
}